// SparseAttention_88837103551208
// MI455X (gfx1250) — compile-verified
//
#include <hip/hip_runtime.h>
#include <math.h>
#include <stdint.h>

typedef __attribute__((ext_vector_type(2))) float v2f;
typedef __attribute__((ext_vector_type(4))) float v4f;
typedef __attribute__((ext_vector_type(8))) float v8f;

#define TT 4096      // sequence length
#define DD 256       // head dim
#define TQ 16        // query tile (WMMA M)
#define WW 32        // key window (2 x WMMA N)
#define WOFF 8       // window starts at q0 - WOFF
#define DP 260       // padded LDS row stride for Q/K (floats)
#define SP 36        // padded LDS row stride for scores/weights (floats)

__global__ __launch_bounds__(32) void band_attn_f32_wmma(
    const float* __restrict__ q, const float* __restrict__ k,
    const float* __restrict__ v, float* __restrict__ out)
{
    __shared__ float Qs[TQ * DP];   // 16.25 KB
    __shared__ float Ks[WW * DP];   // 32.5 KB
    __shared__ float Pa[TQ * SP];   // scores -> softmax weights

    const int lane   = threadIdx.x;        // 0..31 (wave32)
    const int ntiles = TT / TQ;            // 256
    const int bidx   = blockIdx.x / ntiles;
    const int tile   = blockIdx.x % ntiles;
    const int q0     = tile * TQ;
    const int w0     = q0 - WOFF;          // window start (may be <0 or >T-32)

    const float* Qb = q   + (size_t)bidx * TT * DD;
    const float* Kb = k   + (size_t)bidx * TT * DD;
    const float* Vb = v   + (size_t)bidx * TT * DD;
    float*       Ob = out + (size_t)bidx * TT * DD;

    // ---- Stage Q tile (16 x 256) into LDS: async global->LDS (ASYNCcnt) ----
    {
        const uint64_t saddr_q = (uint64_t)(uintptr_t)Qb;
        for (int i = lane; i < TQ * (DD / 4); i += 32) {
            int row = i >> 6;
            int c   = (i & 63) << 2;
            uint32_t goff  = (uint32_t)(((q0 + row) * DD + c) * 4);
            uint32_t laddr = (uint32_t)(uintptr_t)(Qs + row * DP + c);
            asm volatile("global_load_async_to_lds_b128 %0, %1, %2"
                         :: "v"(laddr), "v"(goff), "s"(saddr_q) : "memory");
        }
        // ---- Stage K window (32 x 256), edge rows clamped (masked later) ----
        const uint64_t saddr_k = (uint64_t)(uintptr_t)Kb;
        for (int i = lane; i < WW * (DD / 4); i += 32) {
            int row = i >> 6;
            int c   = (i & 63) << 2;
            int j   = w0 + row;
            j = j < 0 ? 0 : (j > TT - 1 ? TT - 1 : j);
            uint32_t goff  = (uint32_t)((j * DD + c) * 4);
            uint32_t laddr = (uint32_t)(uintptr_t)(Ks + row * DP + c);
            asm volatile("global_load_async_to_lds_b128 %0, %1, %2"
                         :: "v"(laddr), "v"(goff), "s"(saddr_k) : "memory");
        }
    }

    // Prefetch V window rows while the async copies and QK^T run.
    {
        int j = w0 + lane;
        j = j < 0 ? 0 : (j > TT - 1 ? TT - 1 : j);
        __builtin_prefetch(Vb + (size_t)j * DD, 0, 1);
    }

    asm volatile("s_wait_asynccnt 0" ::: "memory");
    __syncthreads();

    // ---- S = Q * K^T over the 32-key window ----
    // A (16x4 f32): lane holds M=lane&15, VGPR v holds K = 2*(lane>>4)+v
    // B (4x16 f32): lane holds N=lane&15, VGPR v holds K = 2*(lane>>4)+v
    const int m    = lane & 15;
    const int koff = (lane >> 4) << 1;      // 0 or 2
    v8f acc0 = {};                          // keys 0..15
    v8f acc1 = {};                          // keys 16..31
    for (int s = 0; s < DD / 4; ++s) {
        int kb = s * 4;
        v2f a  = *(const v2f*)(Qs + m * DP + kb + koff);
        v2f b0 = *(const v2f*)(Ks + m * DP + kb + koff);
        v2f b1 = *(const v2f*)(Ks + (m + 16) * DP + kb + koff);
        acc0 = __builtin_amdgcn_wmma_f32_16x16x4_f32(
            false, a, false, b0, (short)0, acc0, false, false);
        acc1 = __builtin_amdgcn_wmma_f32_16x16x4_f32(
            false, a, false, b1, (short)0, acc1, false, false);
    }

    // ---- Spill scores to LDS (transpose C layout -> row-major) ----
    // C/D layout: lane, VGPR r -> element (M = r + 8*(lane>>4), N = lane&15)
    const int rowbase = (lane >> 4) << 3;   // 0 or 8
    for (int r = 0; r < 8; ++r) {
        Pa[(rowbase + r) * SP + m]      = acc0[r];
        Pa[(rowbase + r) * SP + m + 16] = acc1[r];
    }
    __syncthreads();

    // ---- Band-masked softmax, one row per lane (lanes 0..15) ----
    if (lane < TQ) {
        const int   qi    = q0 + lane;
        const int   lo    = (qi - 5 < 0) ? 0 : qi - 5;
        const int   hi    = (qi + 4 > TT - 1) ? TT - 1 : qi + 4;
        const float scale = 0.0625f;        // 1/sqrt(256)
        float mx = -3.0e38f;
        for (int n = 0; n < WW; ++n) {
            int j = w0 + n;
            if (j >= lo && j <= hi) {
                float s = Pa[lane * SP + n] * scale;
                mx = s > mx ? s : mx;
            }
        }
        float sum = 0.0f;
        for (int n = 0; n < WW; ++n) {
            int j = w0 + n;
            float e = 0.0f;
            if (j >= lo && j <= hi)
                e = __expf(Pa[lane * SP + n] * scale - mx);
            Pa[lane * SP + n] = e;
            sum += e;
        }
        float inv = 1.0f / sum;
        for (int n = 0; n < WW; ++n)
            Pa[lane * SP + n] *= inv;
    }
    __syncthreads();

    // ---- O = P * V : A from LDS weights, B streamed from global V ----
    for (int nb = 0; nb < DD; nb += 16) {
        v8f acc = {};
        for (int s = 0; s < WW / 4; ++s) {
            int k4 = s * 4;
            v2f a = *(const v2f*)(Pa + m * SP + k4 + koff);
            int j0 = w0 + k4 + koff;
            int j1 = j0 + 1;
            int j0c = j0 < 0 ? 0 : (j0 > TT - 1 ? TT - 1 : j0);
            int j1c = j1 < 0 ? 0 : (j1 > TT - 1 ? TT - 1 : j1);
            v2f bb;
            bb.x = Vb[(size_t)j0c * DD + nb + m];
            bb.y = Vb[(size_t)j1c * DD + nb + m];
            acc = __builtin_amdgcn_wmma_f32_16x16x4_f32(
                false, a, false, bb, (short)0, acc, false, false);
        }
        for (int r = 0; r < 8; ++r)
            Ob[(size_t)(q0 + rowbase + r) * DD + nb + m] = acc[r];
    }
}

extern "C" void kernel_launch(void* const* d_in, const int* in_sizes, int n_in,
                              void* d_out, int out_size, void* d_ws, size_t ws_size,
                              hipStream_t stream) {
    (void)n_in; (void)out_size; (void)d_ws; (void)ws_size;
    const float* q = (const float*)d_in[0];
    const float* k = (const float*)d_in[1];
    const float* v = (const float*)d_in[2];
    float* out = (float*)d_out;

    const int b = in_sizes[0] / (TT * DD);   // 4
    dim3 grid(b * (TT / TQ));                // 1024 tiles, one wave32 each
    dim3 block(32);
    band_attn_f32_wmma<<<grid, block, 0, stream>>>(q, k, v, out);
}